// HMMCell_45234595561680
// MI455X (gfx1250) — compile-verified
//
#include <hip/hip_runtime.h>
#include <hip/hip_bf16.h>

typedef __attribute__((ext_vector_type(2))) float v2f;
typedef __attribute__((ext_vector_type(8))) float v8f;

#define BATCH 512
#define T_LEN 1024
#define N_ST  64
#define S_OBS 32

// workspace layout (floats)
#define WS_TRANS 0      // 4096 floats = 2048 v2f  (4 n-tiles x 16 ksteps x 32 lanes)
#define WS_EMIS  4096   // 2048 floats = 1024 v2f  (4 n-tiles x  8 ksteps x 32 lanes)
#define WS_I     6144   // 64 floats

// xor-butterfly sum over each 16-lane half using ds_swizzle_b32 (xor mask is an
// immediate: offset = {xor[14:10], or[9:5]=0, and[4:0]=0x1F})
__device__ __forceinline__ float half16_sum(float v) {
    v += __int_as_float(__builtin_amdgcn_ds_swizzle(__float_as_int(v), 0x041F)); // ^1
    v += __int_as_float(__builtin_amdgcn_ds_swizzle(__float_as_int(v), 0x081F)); // ^2
    v += __int_as_float(__builtin_amdgcn_ds_swizzle(__float_as_int(v), 0x101F)); // ^4
    v += __int_as_float(__builtin_amdgcn_ds_swizzle(__float_as_int(v), 0x201F)); // ^8
    return v;
}

// ---------------------------------------------------------------------------
// Prep: softmax the three parameter tensors and pre-pack WMMA B-operand
// fragments (per-lane v2f for each (n-tile, k-step)) into workspace.
// B-operand layout assumed (ISA 7.12.2 pattern): vgpr j, lanes<16: K=j,
// lanes>=16: K=j+2, N=lane%16  -> per-lane v2f = {K=4k+2h, K=4k+2h+1}.
// ---------------------------------------------------------------------------
__global__ void hmm_prep(const float* __restrict__ emission,
                         const float* __restrict__ transition,
                         const float* __restrict__ initk,
                         float* __restrict__ ws) {
    __shared__ float sA[64][64];
    __shared__ float sB[64][32];
    __shared__ float sI[64];
    const int tid = threadIdx.x;   // 64 threads

    // transition row softmax -> sA
    {
        float m = -3.4e38f;
        for (int j = 0; j < 64; ++j) m = fmaxf(m, transition[tid * 64 + j]);
        float s = 0.f;
        for (int j = 0; j < 64; ++j) { float e = __expf(transition[tid * 64 + j] - m); sA[tid][j] = e; s += e; }
        float r = 1.f / s;
        for (int j = 0; j < 64; ++j) sA[tid][j] *= r;
    }
    // emission row softmax -> sB
    {
        float m = -3.4e38f;
        for (int j = 0; j < 32; ++j) m = fmaxf(m, emission[tid * 32 + j]);
        float s = 0.f;
        for (int j = 0; j < 32; ++j) { float e = __expf(emission[tid * 32 + j] - m); sB[tid][j] = e; s += e; }
        float r = 1.f / s;
        for (int j = 0; j < 32; ++j) sB[tid][j] *= r;
    }
    if (tid == 0) {
        float m = -3.4e38f;
        for (int j = 0; j < 64; ++j) m = fmaxf(m, initk[j]);
        float s = 0.f;
        for (int j = 0; j < 64; ++j) { float e = __expf(initk[j] - m); sI[j] = e; s += e; }
        float r = 1.f / s;
        for (int j = 0; j < 64; ++j) sI[j] *= r;
    }
    __syncthreads();

    // transition fragments: B-operand element (K=i, N=j) = A[i][j]
    for (int idx = tid; idx < 2048; idx += 64) {
        const int lane = idx & 31, frag = idx >> 5;
        const int n = frag >> 4, k = frag & 15;
        const int h = lane >> 4, p = lane & 15;
        ws[WS_TRANS + 2 * idx + 0] = sA[4 * k + 2 * h + 0][16 * n + p];
        ws[WS_TRANS + 2 * idx + 1] = sA[4 * k + 2 * h + 1][16 * n + p];
    }
    // emission fragments: B-operand element (K=s, N=j) = Bm[j][s]
    for (int idx = tid; idx < 1024; idx += 64) {
        const int lane = idx & 31, frag = idx >> 5;
        const int n = frag >> 3, k = frag & 7;
        const int h = lane >> 4, p = lane & 15;
        ws[WS_EMIS + 2 * idx + 0] = sB[16 * n + p][4 * k + 2 * h + 0];
        ws[WS_EMIS + 2 * idx + 1] = sB[16 * n + p][4 * k + 2 * h + 1];
    }
    for (int j = tid; j < 64; j += 64) ws[WS_I + j] = sI[j];
}

// ---------------------------------------------------------------------------
// Fused scan: block = 128 thr = 4 wave32; block owns 16 batch rows,
// wave w owns states [16w, 16w+16). f32 WMMA 16x16x4 for both the emission
// GEMM (K=32) and the transition matvec (K=64).
// ---------------------------------------------------------------------------
__global__ void __launch_bounds__(128) hmm_scan(const float* __restrict__ inputs,
                                                const float* __restrict__ ws,
                                                float* __restrict__ out) {
    __shared__ __align__(16) v2f   s_trans[2048];
    __shared__ __align__(16) v2f   s_emis[1024];
    __shared__ __align__(16) float s_I[64];
    __shared__ __align__(16) float f_lds[16 * 66];       // padded: 66*4B avoids bank conflicts
    __shared__ __align__(16) float inp_lds[2][16 * 34];  // double-buffered 16x32 input tile
    __shared__ __align__(16) float s_part[4][16];        // per-wave partial row sums

    const int tid  = threadIdx.x;
    const int w    = tid >> 5;       // wave id == state N-tile
    const int lane = tid & 31;
    const int h    = lane >> 4;      // lane half (selects K pair / row+8)
    const int p    = lane & 15;
    const int b0   = blockIdx.x * 16;

    // stage packed constants into LDS
    for (int i = tid; i < 2048; i += 128) s_trans[i] = ((const v2f*)(ws + WS_TRANS))[i];
    for (int i = tid; i < 1024; i += 128) s_emis[i]  = ((const v2f*)(ws + WS_EMIS))[i];
    if (tid < 64) s_I[tid] = ws[WS_I + tid];

    // cooperative load of input tile for t=0 (one float4 per thread)
    const int lr = tid >> 3;          // 0..15: batch row in tile
    const int lc = (tid & 7) * 4;     // 0..28: obs column
    const float* __restrict__ inp_ptr = inputs + ((size_t)(b0 + lr) * T_LEN) * S_OBS + lc;
    {
        const float4 v = *(const float4*)inp_ptr;
        inp_lds[0][lr * 34 + lc + 0] = v.x;
        inp_lds[0][lr * 34 + lc + 1] = v.y;
        inp_lds[0][lr * 34 + lc + 2] = v.z;
        inp_lds[0][lr * 34 + lc + 3] = v.w;
    }
    __syncthreads();

    const float Iv = s_I[16 * w + p];
    float ll[8];
    float* outp[8];
#pragma unroll
    for (int r = 0; r < 8; ++r) {
        ll[r] = 0.f;
        // running pointer to out[b0+8h+r][t][16w+p]; bumped by 65 per step
        outp[r] = out + ((size_t)(b0 + 8 * h + r) * T_LEN) * 65 + 16 * w + p;
    }

    for (int t = 0; t < T_LEN; ++t) {
        const int cur = t & 1, nxt = cur ^ 1;

        // prefetch next input tile into registers (overlaps WMMA chain)
        float4 pref;
        const bool has_next = (t + 1 < T_LEN);
        if (has_next)
            pref = *(const float4*)(inp_ptr + (size_t)(t + 1) * S_OBS);
        if (t + 2 < T_LEN)
            __builtin_prefetch(inp_ptr + (size_t)(t + 2) * S_OBS, 0, 1);

        // ---- emission: e = inputs_tile @ Bm^T  (M=16 rows, K=32, this wave's 16 cols)
        v8f e = {0.f, 0.f, 0.f, 0.f, 0.f, 0.f, 0.f, 0.f};
#pragma unroll
        for (int k = 0; k < 8; ++k) {
            v2f a = *(const v2f*)(&inp_lds[cur][p * 34 + 4 * k + 2 * h]);
            v2f b = s_emis[(w * 8 + k) * 32 + lane];
            e = __builtin_amdgcn_wmma_f32_16x16x4_f32(false, a, false, b, (short)0, e,
                                                      false, false);
        }

        // ---- transition: R = fwd @ A  (K=64), or broadcast of I at t==0
        v8f R;
        if (t == 0) {
#pragma unroll
            for (int r = 0; r < 8; ++r) R[r] = Iv;
        } else {
            v8f z = {0.f, 0.f, 0.f, 0.f, 0.f, 0.f, 0.f, 0.f};
            R = z;
#pragma unroll
            for (int k = 0; k < 16; ++k) {
                v2f a = *(const v2f*)(&f_lds[p * 66 + 4 * k + 2 * h]);
                v2f b = s_trans[(w * 16 + k) * 32 + lane];
                R = __builtin_amdgcn_wmma_f32_16x16x4_f32(false, a, false, b, (short)0, R,
                                                          false, false);
            }
        }

        // ---- f = e * R ; per-row partial sums over this wave's 16 states
        float f[8], sp[8];
#pragma unroll
        for (int r = 0; r < 8; ++r) {
            f[r]  = e[r] * R[r];
            sp[r] = half16_sum(f[r]);   // uniform within each 16-lane half
        }
        if (p == 0) {
#pragma unroll
            for (int r = 0; r < 8; ++r) s_part[w][8 * h + r] = sp[r];
        }
        __syncthreads();

        // ---- full row sums, normalize, accumulate loglik
#pragma unroll
        for (int r = 0; r < 8; ++r) {
            const int row = 8 * h + r;
            const float S = s_part[0][row] + s_part[1][row] + s_part[2][row] + s_part[3][row];
            const float inv = 1.0f / S;
            f[r] *= inv;
            ll[r] += __logf(S);
        }

        // ---- emit outputs: out[b][t][0..63] = f, out[b][t][64] = ll
#pragma unroll
        for (int r = 0; r < 8; ++r)
            outp[r][0] = f[r];
        if (w == 0 && p == 0) {
#pragma unroll
            for (int r = 0; r < 8; ++r)
                outp[r][64 - 16 * 0 - 0] = ll[r];   // same row base, column 64 (w==0,p==0)
        }
#pragma unroll
        for (int r = 0; r < 8; ++r)
            outp[r] += 65;

        // ---- stage normalized fwd tile + next input tile for the next step
#pragma unroll
        for (int r = 0; r < 8; ++r)
            f_lds[(8 * h + r) * 66 + 16 * w + p] = f[r];
        if (has_next) {
            inp_lds[nxt][lr * 34 + lc + 0] = pref.x;
            inp_lds[nxt][lr * 34 + lc + 1] = pref.y;
            inp_lds[nxt][lr * 34 + lc + 2] = pref.z;
            inp_lds[nxt][lr * 34 + lc + 3] = pref.w;
        }
        __syncthreads();
    }
}

extern "C" void kernel_launch(void* const* d_in, const int* in_sizes, int n_in,
                              void* d_out, int out_size, void* d_ws, size_t ws_size,
                              hipStream_t stream) {
    const float* inputs     = (const float*)d_in[0];
    const float* emission   = (const float*)d_in[1];
    const float* transition = (const float*)d_in[2];
    const float* initk      = (const float*)d_in[3];
    float* out = (float*)d_out;
    float* ws  = (float*)d_ws;
    (void)in_sizes; (void)n_in; (void)out_size; (void)ws_size;

    hipLaunchKernelGGL(hmm_prep, dim3(1), dim3(64), 0, stream,
                       emission, transition, initk, ws);
    hipLaunchKernelGGL(hmm_scan, dim3(BATCH / 16), dim3(128), 0, stream,
                       inputs, ws, out);
}